// SortPool_8392366096493
// MI455X (gfx1250) — compile-verified
//
#include <hip/hip_runtime.h>
#include <hip/hip_bf16.h>

#define N_NODES   100000
#define N_EDGES   1600000
#define NUM_GRAPHS 2000
#define HIDDEN    64
#define KTOP      7

typedef __attribute__((ext_vector_type(2))) float v2f;
typedef __attribute__((ext_vector_type(8))) float v8f;

// ---------------------------------------------------------------- zero fill
__global__ void zero_f32(float* __restrict__ p, int n) {
    int i = blockIdx.x * blockDim.x + threadIdx.x;
    if (i < n) p[i] = 0.0f;
}

// -------------------------------------------------- edge scatter (mean agg)
// one thread per (edge, feature); 64 features per edge
__global__ void sage_scatter(const float* __restrict__ hin,
                             const int* __restrict__ src,
                             const int* __restrict__ dst,
                             float* __restrict__ msum,
                             float* __restrict__ cnt,
                             int n_edges) {
    long tid = (long)blockIdx.x * blockDim.x + threadIdx.x;
    int e = (int)(tid >> 6);
    int f = (int)(tid & 63);
    if (e >= n_edges) return;
    int s = src[e];
    int d = dst[e];
    atomicAdd(&msum[(long)d * HIDDEN + f], hin[(long)s * HIDDEN + f]);
    if (f == 0) atomicAdd(&cnt[d], 1.0f);
}

// -------------------------------------------------------- SAGE dense stage
// h_out = relu( (msum/max(cnt,1)) @ Wl + b + h_in @ Wr )
// block = 128 threads = 4 waves; block covers 16 node rows x 64 cols,
// wave w covers column tile [16w, 16w+16). K-loop uses V_WMMA_F32_16X16X4_F32.
__global__ __launch_bounds__(128) void sage_gemm(
        const float* __restrict__ msum, const float* __restrict__ cnt,
        const float* __restrict__ hin,
        const float* __restrict__ Wl, const float* __restrict__ bias,
        const float* __restrict__ Wr, float* __restrict__ hout) {
    __shared__ float sMean[16][HIDDEN];
    __shared__ float sX[16][HIDDEN];

    const int rowbase = blockIdx.x * 16;
    const int tid = threadIdx.x;

    for (int i = tid; i < 16 * HIDDEN; i += 128) {
        int r = i >> 6, c = i & 63;
        long node = rowbase + r;
        float cv = cnt[node];
        sMean[r][c] = msum[node * HIDDEN + c] / fmaxf(cv, 1.0f);
        sX[r][c]    = hin [node * HIDDEN + c];
    }
    __syncthreads();

    const int wave    = tid >> 5;       // 0..3 -> column tile
    const int lane    = tid & 31;
    const int colbase = wave * 16;
    const int half    = lane >> 4;      // ISA A/B layout: lanes 16-31 hold K+2,K+3
    const int l16     = lane & 15;      // row (A) / column (B,C,D) within tile

    v8f acc = {};
    // mean @ Wl
    for (int kk = 0; kk < HIDDEN; kk += 4) {
        int k0 = kk + half * 2;
        v2f a, b;
        a.x = sMean[l16][k0];
        a.y = sMean[l16][k0 + 1];
        b.x = Wl[(k0    ) * HIDDEN + colbase + l16];
        b.y = Wl[(k0 + 1) * HIDDEN + colbase + l16];
        acc = __builtin_amdgcn_wmma_f32_16x16x4_f32(false, a, false, b,
                                                    (short)0, acc, false, false);
    }
    // x @ Wr
    for (int kk = 0; kk < HIDDEN; kk += 4) {
        int k0 = kk + half * 2;
        v2f a, b;
        a.x = sX[l16][k0];
        a.y = sX[l16][k0 + 1];
        b.x = Wr[(k0    ) * HIDDEN + colbase + l16];
        b.y = Wr[(k0 + 1) * HIDDEN + colbase + l16];
        acc = __builtin_amdgcn_wmma_f32_16x16x4_f32(false, a, false, b,
                                                    (short)0, acc, false, false);
    }

    const int col = colbase + l16;
    const float bv = bias[col];
#pragma unroll
    for (int v = 0; v < 8; ++v) {
        int row = v + half * 8;         // C/D layout: VGPR v -> M = v (+8 for hi half)
        float val = fmaxf(acc[v] + bv, 0.0f);
        hout[(long)(rowbase + row) * HIDDEN + col] = val;
    }
}

// ----------------------------------------------------------------- sortpool
// one 64-thread block per graph; batch[] is sorted so nodes are contiguous.
// 7 rounds of stable max-selection on channel 63 (desc value, asc index).
__global__ __launch_bounds__(64) void sortpool_kernel(
        const float* __restrict__ h, const int* __restrict__ batch,
        float* __restrict__ gout, int n_nodes) {
    const int g = blockIdx.x;
    const int tid = threadIdx.x;
    __shared__ float sval[64];
    __shared__ int   sidx[64];
    __shared__ int   ssel[KTOP];
    __shared__ int   sstart, send;

    if (tid == 0) {
        int lo = 0, hi = n_nodes;
        while (lo < hi) { int mid = (lo + hi) >> 1; if (batch[mid] < g) lo = mid + 1; else hi = mid; }
        sstart = lo;
        hi = n_nodes;
        while (lo < hi) { int mid = (lo + hi) >> 1; if (batch[mid] <= g) lo = mid + 1; else hi = mid; }
        send = lo;
    }
    __syncthreads();
    const int start = sstart, end = send;

    float prevVal = 3.4e38f;
    int   prevIdx = -1;
    for (int k = 0; k < KTOP; ++k) {
        float bestV = -3.4e38f;
        int   bestI = 0x7FFFFFFF;
        for (int i = start + tid; i < end; i += 64) {
            float v = h[(long)i * HIDDEN + 63];
            bool after = (v < prevVal) || (v == prevVal && i > prevIdx);
            if (after && (v > bestV || (v == bestV && i < bestI))) { bestV = v; bestI = i; }
        }
        sval[tid] = bestV;
        sidx[tid] = bestI;
        __syncthreads();
        for (int s = 32; s > 0; s >>= 1) {
            if (tid < s) {
                float v2 = sval[tid + s]; int i2 = sidx[tid + s];
                if (v2 > sval[tid] || (v2 == sval[tid] && i2 < sidx[tid])) {
                    sval[tid] = v2; sidx[tid] = i2;
                }
            }
            __syncthreads();
        }
        if (tid == 0) ssel[k] = (sidx[0] == 0x7FFFFFFF) ? -1 : sidx[0];
        __syncthreads();
        prevVal = sval[0];
        prevIdx = sidx[0];
        __syncthreads();
    }
    for (int k = 0; k < KTOP; ++k) {
        int sel = ssel[k];
        float v = (sel >= 0) ? h[(long)sel * HIDDEN + tid] : 0.0f;
        gout[((long)g * KTOP + k) * HIDDEN + tid] = v;
    }
}

// ------------------------------------------------------------------- head
// per graph: conv1d(k=3, 64->32) + relu, flatten, lin1 + relu, lin2,
// log_softmax.  one 64-thread block per graph.
__global__ __launch_bounds__(64) void head_kernel(
        const float* __restrict__ gpool,
        const float* __restrict__ cw, const float* __restrict__ cb,
        const float* __restrict__ l1w, const float* __restrict__ l1b,
        const float* __restrict__ l2w, const float* __restrict__ l2b,
        float* __restrict__ out) {
    const int g = blockIdx.x;
    const int tid = threadIdx.x;
    __shared__ float sg[KTOP * HIDDEN];   // [k][h]
    __shared__ float flat[160];           // [o*5 + t]
    __shared__ float sh1[HIDDEN];
    __shared__ float slog[10];

    for (int i = tid; i < KTOP * HIDDEN; i += 64)
        sg[i] = gpool[(long)g * KTOP * HIDDEN + i];
    __syncthreads();

    // conv output c[o][t] = relu( b[o] + sum_{s,i} g[t+s][i] * w[o][i][s] )
    for (int m = tid; m < 160; m += 64) {
        int o = m / 5, t = m % 5;
        float acc = cb[o];
        for (int s = 0; s < 3; ++s)
            for (int i = 0; i < HIDDEN; ++i)
                acc += sg[(t + s) * HIDDEN + i] * cw[(o * HIDDEN + i) * 3 + s];
        flat[m] = fmaxf(acc, 0.0f);
    }
    __syncthreads();

    {   // lin1: 160 -> 64, thread tid computes output column tid
        float acc = l1b[tid];
        for (int m = 0; m < 160; ++m) acc += flat[m] * l1w[m * HIDDEN + tid];
        sh1[tid] = fmaxf(acc, 0.0f);
    }
    __syncthreads();

    if (tid < 10) {   // lin2: 64 -> 10
        float acc = l2b[tid];
        for (int j = 0; j < HIDDEN; ++j) acc += sh1[j] * l2w[j * 10 + tid];
        slog[tid] = acc;
    }
    __syncthreads();

    if (tid == 0) {   // log_softmax over 10 classes
        float mx = slog[0];
        for (int c = 1; c < 10; ++c) mx = fmaxf(mx, slog[c]);
        float se = 0.0f;
        for (int c = 0; c < 10; ++c) se += expf(slog[c] - mx);
        float lse = logf(se) + mx;
        for (int c = 0; c < 10; ++c) out[(long)g * 10 + c] = slog[c] - lse;
    }
}

// ------------------------------------------------------------------ driver
extern "C" void kernel_launch(void* const* d_in, const int* in_sizes, int n_in,
                              void* d_out, int out_size, void* d_ws, size_t ws_size,
                              hipStream_t stream) {
    const float* x     = (const float*)d_in[0];
    const int*   ei    = (const int*)  d_in[1];
    const int*   src   = ei;
    const int*   dst   = ei + N_EDGES;
    const int*   batch = (const int*)  d_in[2];
    const float* W1l = (const float*)d_in[3];
    const float* b1  = (const float*)d_in[4];
    const float* W1r = (const float*)d_in[5];
    const float* W2l = (const float*)d_in[6];
    const float* b2  = (const float*)d_in[7];
    const float* W2r = (const float*)d_in[8];
    const float* W3l = (const float*)d_in[9];
    const float* b3  = (const float*)d_in[10];
    const float* W3r = (const float*)d_in[11];
    const float* cw  = (const float*)d_in[12];
    const float* cb  = (const float*)d_in[13];
    const float* l1w = (const float*)d_in[14];
    const float* l1b = (const float*)d_in[15];
    const float* l2w = (const float*)d_in[16];
    const float* l2b = (const float*)d_in[17];

    // workspace layout (f32): msum | cnt | hA | hB | gpool
    float* msum  = (float*)d_ws;
    float* cnt   = msum + (long)N_NODES * HIDDEN;
    float* hA    = cnt  + N_NODES;
    float* hB    = hA   + (long)N_NODES * HIDDEN;
    float* gpool = hB   + (long)N_NODES * HIDDEN;

    const int  nzero     = N_NODES * HIDDEN + N_NODES;   // msum + cnt contiguous
    const long nscatter  = (long)N_EDGES * HIDDEN;
    const int  sc_blocks = (int)((nscatter + 255) / 256);
    const int  zb        = (nzero + 255) / 256;
    const int  gemm_blk  = N_NODES / 16;                 // 6250 exactly

    // layer 1
    zero_f32<<<zb, 256, 0, stream>>>(msum, nzero);
    sage_scatter<<<sc_blocks, 256, 0, stream>>>(x, src, dst, msum, cnt, N_EDGES);
    sage_gemm<<<gemm_blk, 128, 0, stream>>>(msum, cnt, x, W1l, b1, W1r, hA);
    // layer 2
    zero_f32<<<zb, 256, 0, stream>>>(msum, nzero);
    sage_scatter<<<sc_blocks, 256, 0, stream>>>(hA, src, dst, msum, cnt, N_EDGES);
    sage_gemm<<<gemm_blk, 128, 0, stream>>>(msum, cnt, hA, W2l, b2, W2r, hB);
    // layer 3
    zero_f32<<<zb, 256, 0, stream>>>(msum, nzero);
    sage_scatter<<<sc_blocks, 256, 0, stream>>>(hB, src, dst, msum, cnt, N_EDGES);
    sage_gemm<<<gemm_blk, 128, 0, stream>>>(msum, cnt, hB, W3l, b3, W3r, hA);

    sortpool_kernel<<<NUM_GRAPHS, 64, 0, stream>>>(hA, batch, gpool, N_NODES);
    head_kernel<<<NUM_GRAPHS, 64, 0, stream>>>(gpool, cw, cb, l1w, l1b, l2w, l2b,
                                               (float*)d_out);
}